// ParameterizedGCN_10153302687991
// MI455X (gfx1250) — compile-verified
//
#include <hip/hip_runtime.h>
#include <hip/hip_bf16.h>

#define N_NODES 50000
#define N_FEAT  256
#define N_HID   256
#define N_CLASS 40
#define N_EDGES 800000

typedef float v2f __attribute__((ext_vector_type(2)));
typedef float v8f __attribute__((ext_vector_type(8)));

// Native L2 float atomic-add (no return) at device scope.
// ISA: GLOBAL_ATOMIC_ADD_F32 (op 86), RMW executes at the L2 atomic units.
__device__ __forceinline__ void atomAddF(float* p, float v) {
    asm volatile("global_atomic_add_f32 %0, %1, off scope:SCOPE_DEV"
                 :: "v"(p), "v"(v) : "memory");
}

// ---------------------------------------------------------------- utilities
__global__ __launch_bounds__(256) void zero_f32(float* p, int n) {
    int i = blockIdx.x * 256 + threadIdx.x;
    if (i < n) p[i] = 0.0f;
}

__global__ __launch_bounds__(256) void count_deg(const long long* __restrict__ dst,
                                                 float* __restrict__ deg, int E) {
    int e = blockIdx.x * 256 + threadIdx.x;
    if (e < E) atomAddF(deg + dst[e], 1.0f);
}

// dinv[i] = rsqrt(deg_from_edges[i] + 1)   (+1 = self loop)
__global__ __launch_bounds__(256) void make_dinv(float* deg, int n) {
    int i = blockIdx.x * 256 + threadIdx.x;
    if (i < n) deg[i] = rsqrtf(deg[i] + 1.0f);
}

// ---------------------------------------------------------------- GEMM (f32 WMMA)
// H[M,N] = X[M,256] @ W[256,N].  One wave per 16x16 output tile.
// M must be a multiple of 16 (50000 = 3125*16).  N may be ragged (guarded).
__global__ __launch_bounds__(32) void gemm_wmma_f32(const float* __restrict__ X,
                                                    const float* __restrict__ W,
                                                    float* __restrict__ H,
                                                    int N, int Ntiles) {
    const int K = 256;
    __shared__ float lds_w[K * 16];   // 16 KB: W tile, row-major [K][16]

    int tile = blockIdx.x;
    int m0 = (tile / Ntiles) * 16;
    int n0 = (tile % Ntiles) * 16;
    int lane = threadIdx.x & 31;

    // cooperative load of W[:, n0:n0+16) into LDS (zeros past N)
    for (int i = lane; i < K * 16; i += 32) {
        int kk = i >> 4;
        int c  = i & 15;
        lds_w[i] = (n0 + c < N) ? W[kk * N + n0 + c] : 0.0f;
    }
    __syncthreads();

    const int half = lane >> 4;    // 0: K pair {0,1}, 1: K pair {2,3}
    const int lr   = lane & 15;
    const float* xrow = X + (long long)(m0 + lr) * K;

    v8f acc = {};
    #pragma unroll 4
    for (int k = 0; k < K; k += 4) {
        int ks = k + 2 * half;
        v2f a, b;
        a[0] = xrow[ks];
        a[1] = xrow[ks + 1];
        b[0] = lds_w[ks * 16 + lr];
        b[1] = lds_w[(ks + 1) * 16 + lr];
        // D = A(16x4) * B(4x16) + C ; f32 in, f32 accumulate
        acc = __builtin_amdgcn_wmma_f32_16x16x4_f32(false, a, false, b,
                                                    (short)0, acc, false, false);
    }

    // C/D layout: VGPR r -> lanes 0-15: M=r, N=lane ; lanes 16-31: M=8+r, N=lane-16
    int col = n0 + lr;
    if (col < N) {
        int rbase = m0 + 8 * half;
        #pragma unroll
        for (int r = 0; r < 8; ++r)
            H[(long long)(rbase + r) * N + col] = acc[r];
    }
}

// ---------------------------------------------------------------- aggregation
// One block per node row: agg[row,f] = dinv[row]^2 * h[row,f]
// (self-loop term; also fully initializes agg, overwriting poison)
__global__ __launch_bounds__(256) void self_init(const float* __restrict__ h,
                                                 const float* __restrict__ dinv,
                                                 float* __restrict__ agg, int N) {
    int row = blockIdx.x;
    float di = dinv[row];
    float sc = di * di;
    const float* hr = h + (long long)row * N;
    float*       ar = agg + (long long)row * N;
    for (int f = threadIdx.x; f < N; f += 256)
        ar[f] = sc * hr[f];
}

// wave-per-edge: agg[dst,f] += dinv[src]*dinv[dst]*h[src,f]  (float4 gather)
__global__ __launch_bounds__(256) void edge_agg(const float* __restrict__ h,
                                                const long long* __restrict__ src,
                                                const long long* __restrict__ dst,
                                                const float* __restrict__ dinv,
                                                float* __restrict__ agg,
                                                int E, int N) {
    int wave = (blockIdx.x * 256 + threadIdx.x) >> 5;
    int lane = threadIdx.x & 31;
    if (wave >= E) return;
    long long s = src[wave];
    long long d = dst[wave];
    float nrm = dinv[s] * dinv[d];
    const float4* hs4 = (const float4*)(h + s * N);
    float*        ad  = agg + d * N;
    int nq = N >> 2;                       // N is a multiple of 4 (256 or 40)
    for (int q = lane; q < nq; q += 32) {
        float4 hv = hs4[q];                // global_load_b128, coalesced
        float* ap = ad + q * 4;
        atomAddF(ap + 0, nrm * hv.x);
        atomAddF(ap + 1, nrm * hv.y);
        atomAddF(ap + 2, nrm * hv.z);
        atomAddF(ap + 3, nrm * hv.w);
    }
}

// One block per node row: out[row,f] = (agg[row,f] + b[f]), optional ReLU
__global__ __launch_bounds__(256) void bias_act(const float* __restrict__ agg,
                                                const float* __restrict__ b,
                                                float* __restrict__ out,
                                                int N, int do_relu) {
    int row = blockIdx.x;
    const float* ar = agg + (long long)row * N;
    float*       orow = out + (long long)row * N;
    for (int f = threadIdx.x; f < N; f += 256) {
        float v = ar[f] + b[f];
        orow[f] = (do_relu && v < 0.0f) ? 0.0f : v;
    }
}

// ---------------------------------------------------------------- launcher
extern "C" void kernel_launch(void* const* d_in, const int* in_sizes, int n_in,
                              void* d_out, int out_size, void* d_ws, size_t ws_size,
                              hipStream_t stream) {
    const float*     x  = (const float*)d_in[0];
    const long long* ei = (const long long*)d_in[1];   // int64 [2, E]
    const float* W1 = (const float*)d_in[2];
    const float* b1 = (const float*)d_in[3];
    const float* W2 = (const float*)d_in[4];
    const float* b2 = (const float*)d_in[5];
    const float* W3 = (const float*)d_in[6];
    const float* b3 = (const float*)d_in[7];
    const float* W4 = (const float*)d_in[8];
    const float* b4 = (const float*)d_in[9];
    float* outf = (float*)d_out;

    const long long* esrc = ei;
    const long long* edst = ei + N_EDGES;

    float* wsf  = (float*)d_ws;
    float* hbuf = wsf;                                   // 50000*256
    float* agg  = wsf + (size_t)N_NODES * N_HID;         // 50000*256
    float* dinv = agg + (size_t)N_NODES * N_HID;         // 50000

    const int MT = N_NODES / 16;          // 3125
    const int edgeBlocks = (N_EDGES * 32 + 255) / 256;

    // --- normalization factors ---
    zero_f32<<<(N_NODES + 255) / 256, 256, 0, stream>>>(dinv, N_NODES);
    count_deg<<<(N_EDGES + 255) / 256, 256, 0, stream>>>(edst, dinv, N_EDGES);
    make_dinv<<<(N_NODES + 255) / 256, 256, 0, stream>>>(dinv, N_NODES);

    // --- layer 1 ---
    gemm_wmma_f32<<<MT * 16, 32, 0, stream>>>(x, W1, hbuf, N_HID, 16);
    self_init<<<N_NODES, 256, 0, stream>>>(hbuf, dinv, agg, N_HID);
    edge_agg<<<edgeBlocks, 256, 0, stream>>>(hbuf, esrc, edst, dinv, agg, N_EDGES, N_HID);
    bias_act<<<N_NODES, 256, 0, stream>>>(agg, b1, agg, N_HID, 1);

    // --- layer 2 ---
    gemm_wmma_f32<<<MT * 16, 32, 0, stream>>>(agg, W2, hbuf, N_HID, 16);
    self_init<<<N_NODES, 256, 0, stream>>>(hbuf, dinv, agg, N_HID);
    edge_agg<<<edgeBlocks, 256, 0, stream>>>(hbuf, esrc, edst, dinv, agg, N_EDGES, N_HID);
    bias_act<<<N_NODES, 256, 0, stream>>>(agg, b2, agg, N_HID, 1);

    // --- layer 3 ---
    gemm_wmma_f32<<<MT * 16, 32, 0, stream>>>(agg, W3, hbuf, N_HID, 16);
    self_init<<<N_NODES, 256, 0, stream>>>(hbuf, dinv, agg, N_HID);
    edge_agg<<<edgeBlocks, 256, 0, stream>>>(hbuf, esrc, edst, dinv, agg, N_EDGES, N_HID);
    bias_act<<<N_NODES, 256, 0, stream>>>(agg, b3, agg, N_HID, 1);

    // --- layer 4 (N=40, no ReLU, accumulate directly in d_out) ---
    gemm_wmma_f32<<<MT * 3, 32, 0, stream>>>(agg, W4, hbuf, N_CLASS, 3);
    self_init<<<N_NODES, 256, 0, stream>>>(hbuf, dinv, outf, N_CLASS);
    edge_agg<<<edgeBlocks, 256, 0, stream>>>(hbuf, esrc, edst, dinv, outf, N_EDGES, N_CLASS);
    bias_act<<<N_NODES, 256, 0, stream>>>(outf, b4, outf, N_CLASS, 0);
}